// EDTLoss_66271345377303
// MI455X (gfx1250) — compile-verified
//
#include <hip/hip_runtime.h>
#include <hip/hip_bf16.h>

#define B_   16
#define C_   4
#define H_   320
#define W_   320
#define HW_  (H_ * W_)
#define BIG_ 1.0e6f

#define NPIX    (B_ * HW_)        // 1,638,400 pixels

#define K3_BLOCKS  256
#define K3_THREADS 256            // 8 waves/block
#define WAVES_TOTAL     (K3_BLOCKS * (K3_THREADS / 32))  // 2048
#define PIX_PER_WAVE    (NPIX / WAVES_TOTAL)             // 800 (divides HW_)
#define CHUNKS_PER_WAVE (PIX_PER_WAVE / 16)              // 50 (compile-time)

typedef float v2f __attribute__((ext_vector_type(2)));
typedef float v8f __attribute__((ext_vector_type(8)));

// v_min3_num_f32: single-instruction 3-way float min (CDNA5 "_num" naming).
__device__ __forceinline__ float min3f(float a, float b, float c) {
    float d;
    asm("v_min3_num_f32 %0, %1, %2, %3" : "=v"(d) : "v"(a), "v"(b), "v"(c));
    return d;
}

// ---------------------------------------------------------------- k0: init
__global__ void k0_init(float* __restrict__ dmax) {
    if (threadIdx.x < B_ * C_) dmax[threadIdx.x] = 0.0f;
}

// ------------------------------------------------- k1: 1D pass along H (cols)
// One block per (b,c) plane, one thread per column w. Forward scan stores
// d_fwd, backward scan combines: g = min(d_fwd, d_bwd, BIG); stores g^2.
__global__ __launch_bounds__(W_) void k1_cols(const int* __restrict__ targets,
                                              float* __restrict__ buf) {
    const int bc = blockIdx.x;
    const int b  = bc >> 2;
    const int c  = bc & 3;
    const int w  = threadIdx.x;
    const int* tg = targets + b * HW_;
    float* g = buf + (size_t)bc * HW_;

    float last = -BIG_;
    for (int h = 0; h < H_; ++h) {
        if (tg[h * W_ + w] == c) last = (float)h;
        g[h * W_ + w] = (float)h - last;               // d_fwd
    }
    float nxt = BIG_;
    for (int h = H_ - 1; h >= 0; --h) {
        if (tg[h * W_ + w] == c) nxt = (float)h;
        float dfwd = g[h * W_ + w];
        float d = min3f(dfwd, nxt - (float)h, BIG_);
        g[h * W_ + w] = d * d;                         // g^2
    }
}

// ----------------------------------- k2: brute-force 1D parabola pass along W
// One block per row (B*C*H rows). Row g2 cached in LDS; each thread j does
// min_k g2[k] + (j-k)^2, writes dist=sqrt in place, block-max -> atomicMax.
// Two independent min3 accumulator chains -> dependency depth 1 per iter.
__global__ __launch_bounds__(W_) void k2_rows(float* __restrict__ buf,
                                              float* __restrict__ dmax) {
    __shared__ __align__(16) float srow[W_];
    __shared__ float swmax[W_ / 32];

    const int row = blockIdx.x;              // [0, B*C*H)
    const int bc  = row / H_;
    float* g = buf + (size_t)row * W_;
    const int j = threadIdx.x;

    srow[j] = g[j];
    __syncthreads();

    float m0 = 3.0e38f, m1 = 3.0e38f;
    const float fj = (float)j;
    const float4* s4 = (const float4*)srow;
#pragma unroll 4
    for (int k = 0; k < W_; k += 4) {
        float4 q = s4[k >> 2];
        float d0 = fj - (float)k;
        float d1 = d0 - 1.0f, d2 = d0 - 2.0f, d3 = d0 - 3.0f;
        float c0 = fmaf(d0, d0, q.x);
        float c1 = fmaf(d1, d1, q.y);
        float c2 = fmaf(d2, d2, q.z);
        float c3 = fmaf(d3, d3, q.w);
        m0 = min3f(m0, c0, c1);               // v_min3_num_f32
        m1 = min3f(m1, c2, c3);               // v_min3_num_f32
    }
    float m = fminf(m0, m1);
    float dist = __builtin_amdgcn_sqrtf(m);   // single v_sqrt_f32
    g[j] = dist;                              // in place: everyone read LDS copy

    // block max (exact max => order-independent => deterministic)
    float mx = dist;
    for (int s = 16; s > 0; s >>= 1) mx = fmaxf(mx, __shfl_xor(mx, s, 32));
    if ((j & 31) == 0) swmax[j >> 5] = mx;
    __syncthreads();
    if (j == 0) {
        float bm = swmax[0];
        for (int i = 1; i < W_ / 32; ++i) bm = fmaxf(bm, swmax[i]);
        atomicMax((unsigned int*)&dmax[bc], __float_as_uint(bm)); // dist >= 0
    }
}

// --------------------------- k2b: dmax -> scale, hoisted out of the hot loop
// scale = keep/(dmax+1e-8); keep == (class>0 && class present). Absent class
// => dist ~ BIG=1e6 => dmax ~ 1e6, so "has_fg" <=> dmax < 1e5.
__global__ void k2b_scale(float* __restrict__ dmax) {
    const int i = threadIdx.x;
    if (i < B_ * C_) {
        float dm = dmax[i];
        int   c  = i & 3;
        dmax[i] = (c > 0 && dm < 1.0e5f) ? __builtin_amdgcn_rcpf(dm + 1e-8f)
                                         : 0.0f;
    }
}

// ------------------------- k3: softmax * normalized dist, WMMA f32 16x16x4
// A[m][k] = probs(pixel m, class k); B[k][n] = dist(class k, pixel n)*scale.
// diag(C) accumulates per-pixel sum over classes. K=4 == C_ exactly.
// Static partition: 2048 waves x 50 chunks, each wave inside ONE batch
// (PIX_PER_WAVE=800 divides HW_) -> no divides, no divergence, EXEC all-1s.
__global__ __launch_bounds__(K3_THREADS) void k3_wmma(const float* __restrict__ preds,
                                                      const float* __restrict__ dist,
                                                      const float* __restrict__ scale,
                                                      float* __restrict__ partials) {
    __shared__ float swsum[K3_THREADS / 32];
    const int lane  = threadIdx.x & 31;
    const int wid   = threadIdx.x >> 5;
    const int gw    = blockIdx.x * (K3_THREADS / 32) + wid;
    const int mslot = lane & 15;
    const int k0    = (lane < 16) ? 0 : 2;     // classes this lane supplies

    const int pbase  = gw * PIX_PER_WAVE;      // entire range in one batch
    const int b      = pbase / HW_;            // one division, outside loop
    const int hwbase = (pbase - b * HW_) + mslot;

    const float* prb = preds + (size_t)(b * C_) * HW_;
    const float* dsb = dist  + (size_t)(b * C_) * HW_ + (size_t)k0 * HW_;
    const float  s0  = scale[b * C_ + k0];
    const float  s1  = scale[b * C_ + k0 + 1];

    v8f acc = {0.f, 0.f, 0.f, 0.f, 0.f, 0.f, 0.f, 0.f};

#pragma unroll 2
    for (int i = 0; i < CHUNKS_PER_WAVE; ++i) {   // compile-time trip count
        const int hw = hwbase + i * 16;
        float x0 = prb[hw];
        float x1 = prb[hw + HW_];
        float x2 = prb[hw + 2 * HW_];
        float x3 = prb[hw + 3 * HW_];
        float mx = fmaxf(fmaxf(x0, x1), fmaxf(x2, x3));
        float e0 = __expf(x0 - mx), e1 = __expf(x1 - mx);
        float e2 = __expf(x2 - mx), e3 = __expf(x3 - mx);
        float inv = __builtin_amdgcn_rcpf(e0 + e1 + e2 + e3); // v_rcp_f32

        v2f a, bb;
        a.x = ((k0 == 0) ? e0 : e2) * inv;     // A[m][k0]
        a.y = ((k0 == 0) ? e1 : e3) * inv;     // A[m][k0+1]
        bb.x = dsb[hw]       * s0;             // B[k0  ][m]
        bb.y = dsb[hw + HW_] * s1;             // B[k0+1][m]

        acc = __builtin_amdgcn_wmma_f32_16x16x4_f32(
            /*neg_a=*/false, a, /*neg_b=*/false, bb,
            /*c_mod=*/(short)0, acc, /*reuse_a=*/false, /*reuse_b=*/false);
    }

    // diagonal (m,m): m<8 -> lane m, VGPR m ; m>=8 -> N=lane-16, M=r+8
    // => r = lane-24, valid lanes 24..31.
    int r = (lane < 8) ? lane : (lane - 24);
    float sel = acc[0];
    sel = (r == 1) ? acc[1] : sel;
    sel = (r == 2) ? acc[2] : sel;
    sel = (r == 3) ? acc[3] : sel;
    sel = (r == 4) ? acc[4] : sel;
    sel = (r == 5) ? acc[5] : sel;
    sel = (r == 6) ? acc[6] : sel;
    sel = (r == 7) ? acc[7] : sel;
    float dv = ((lane < 8) || (lane >= 24)) ? sel : 0.0f;

    for (int s = 16; s > 0; s >>= 1) dv += __shfl_xor(dv, s, 32);
    if (lane == 0) swsum[wid] = dv;
    __syncthreads();
    if (threadIdx.x == 0) {
        float t = 0.0f;
        for (int i = 0; i < K3_THREADS / 32; ++i) t += swsum[i];
        partials[blockIdx.x] = t;             // deterministic per-block partial
    }
}

// ---------------------------------------------------------- k4: final reduce
__global__ __launch_bounds__(K3_BLOCKS) void k4_final(const float* __restrict__ partials,
                                                      float* __restrict__ out) {
    __shared__ float s[K3_BLOCKS];
    const int t = threadIdx.x;
    s[t] = partials[t];
    __syncthreads();
    for (int stp = K3_BLOCKS / 2; stp > 0; stp >>= 1) {
        if (t < stp) s[t] += s[t + stp];
        __syncthreads();
    }
    if (t == 0) out[0] = s[0] * (1.0f / (float)(B_ * C_));
}

// ---------------------------------------------------------------------------
extern "C" void kernel_launch(void* const* d_in, const int* in_sizes, int n_in,
                              void* d_out, int out_size, void* d_ws, size_t ws_size,
                              hipStream_t stream) {
    const float* preds   = (const float*)d_in[0];   // [B,C,H,W] f32
    const int*   targets = (const int*)d_in[1];     // [B,H,W] int

    float* ws       = (float*)d_ws;
    float* dmax     = ws;                            // 64 floats (256 B aligned)
    float* buf      = ws + 64;                       // B*C*H*W floats (~26.2 MB)
    float* partials = buf + (size_t)B_ * C_ * HW_;   // K3_BLOCKS floats

    k0_init  <<<1, 64, 0, stream>>>(dmax);
    k1_cols  <<<B_ * C_, W_, 0, stream>>>(targets, buf);
    k2_rows  <<<B_ * C_ * H_, W_, 0, stream>>>(buf, dmax);
    k2b_scale<<<1, 64, 0, stream>>>(dmax);           // dmax -> scale, in place
    k3_wmma  <<<K3_BLOCKS, K3_THREADS, 0, stream>>>(preds, buf, dmax, partials);
    k4_final <<<1, K3_BLOCKS, 0, stream>>>(partials, (float*)d_out);
}